// Multiple_Head_RegionProposalNetwork_2920577761472
// MI455X (gfx1250) — compile-verified
//
#include <hip/hip_runtime.h>
#include <math.h>

// ---------------- problem constants ----------------
#define NLVL   5
#define AANCH  3
#define BATCH  8
#define POST   1000
#define M_CAND 4768            // 1000+1000+1000+1000+768
#define SORTN  8192            // next pow2 >= M_CAND
#define IMG_SZ 1024.0f
#define LVL_OFF 1025.0f        // IMG + 1 (batched_nms level offset)
#define BBOX_CLIP_F 4.135166556742356f
#define NMS_TH_F 0.7f

// workspace layout (bytes)
#define WS_SCORE 0                           // 8*4768 f32   = 152576
#define WS_BOX   152576                      // 8*4768*4 f32 = 610304
#define WS_THR   762880                      // 40 u32
#define WS_CNTG  763040                      // 40 i32
#define WS_CTRG  763200                      // 40 i32
#define WS_CTRE  763360                      // 40 i32

__device__ __forceinline__ unsigned sort_key(float x) {
    unsigned u = __float_as_uint(x);
    return u ^ ((u >> 31) ? 0xFFFFFFFFu : 0x80000000u);   // monotonic: bigger float -> bigger key
}

// ---------------- K1: per (batch, level) k-th largest logit via 4-round radix select ----------------
__global__ __launch_bounds__(1024)
void rpn_topk_thresh(const float* s0, const float* s1, const float* s2, const float* s3, const float* s4,
                     unsigned* thr, int* cntg, int* ctrg, int* ctre) {
    const int blk = blockIdx.x;
    const int b = blk / NLVL;
    const int l = blk % NLVL;
    const float* sp; int H;
    switch (l) {
        case 0: sp = s0; H = 256; break;
        case 1: sp = s1; H = 128; break;
        case 2: sp = s2; H =  64; break;
        case 3: sp = s3; H =  32; break;
        default: sp = s4; H = 16; break;
    }
    const int n = H * H * AANCH;
    const int k = (n < POST) ? n : POST;
    const float* base = sp + (size_t)b * n;

    __shared__ unsigned s_hist[256];
    __shared__ unsigned s_prefix;
    __shared__ int s_want;
    if (threadIdx.x == 0) { s_prefix = 0u; s_want = k; }
    __syncthreads();

    for (int round = 0; round < 4; ++round) {
        const int shift = 24 - 8 * round;
        for (int i = threadIdx.x; i < 256; i += blockDim.x) s_hist[i] = 0u;
        __syncthreads();
        const unsigned prefix = s_prefix;
        const unsigned mask_high = (round == 0) ? 0u : (0xFFFFFFFFu << (shift + 8));
        for (int i = threadIdx.x; i < n; i += blockDim.x) {
            const unsigned key = sort_key(base[i]);
            if ((key & mask_high) == (prefix & mask_high))
                atomicAdd(&s_hist[(key >> shift) & 255u], 1u);
        }
        __syncthreads();
        if (threadIdx.x == 0) {
            int want = s_want;
            unsigned cum = 0;
            int bin = 0;
            for (int bb = 255; bb >= 0; --bb) {
                const unsigned c = s_hist[bb];
                if ((int)(cum + c) >= want) { bin = bb; break; }
                cum += c;
            }
            s_prefix = prefix | ((unsigned)bin << shift);
            s_want = want - (int)cum;
        }
        __syncthreads();
    }
    if (threadIdx.x == 0) {
        thr[blk]  = s_prefix;         // k-th largest key
        cntg[blk] = k - s_want;       // count of keys strictly greater than threshold
        ctrg[blk] = 0;
        ctre[blk] = 0;
    }
}

// ---------------- K2: compact top-k, decode, clip, sigmoid ----------------
__global__ __launch_bounds__(256)
void rpn_select_decode(const float* __restrict__ scores, const float* __restrict__ deltas,
                       const float* __restrict__ anchors,
                       int H, int n, int k, int lvl, int ck, int aoff,
                       const unsigned* __restrict__ thr, const int* __restrict__ cntg,
                       int* ctrg, int* ctre,
                       float* __restrict__ cand_score, float* __restrict__ cand_box) {
    const int b = blockIdx.y;
    const int i = blockIdx.x * blockDim.x + threadIdx.x;
    if (i >= n) return;
    const int blk = b * NLVL + lvl;

    // permuted flat index i -> (h, w, a)
    const int a  = i % AANCH;
    const int hw = i / AANCH;
    const int w  = hw % H;
    const int h  = hw / H;

    const float logit = scores[(((size_t)b * AANCH + a) * H + h) * H + w];
    const unsigned key = sort_key(logit);
    const unsigned T = thr[blk];
    const int cg = cntg[blk];

    int slot = -1;
    if (key > T) {
        slot = atomicAdd(&ctrg[blk], 1);
    } else if (key == T) {
        const int q = atomicAdd(&ctre[blk], 1);
        if (q < k - cg) slot = cg + q;
    }
    if (slot < 0) return;

    // decode (BoxCoder weights (1,1,1,1))
    const float* anc = anchors + (size_t)(aoff + i) * 4;
    const float ax1 = anc[0], ay1 = anc[1], ax2 = anc[2], ay2 = anc[3];
    const float aw = ax2 - ax1, ah = ay2 - ay1;
    const float cx = ax1 + 0.5f * aw, cy = ay1 + 0.5f * ah;

    const size_t hwstride = (size_t)H * H;
    const float* dbase = deltas + ((size_t)b * (4 * AANCH) + a * 4) * hwstride + (size_t)h * H + w;
    const float dx = dbase[0];
    const float dy = dbase[hwstride];
    const float dw = fminf(dbase[2 * hwstride], BBOX_CLIP_F);
    const float dh = fminf(dbase[3 * hwstride], BBOX_CLIP_F);

    const float pcx = dx * aw + cx, pcy = dy * ah + cy;
    const float pw = expf(dw) * aw, ph = expf(dh) * ah;
    float x1 = pcx - 0.5f * pw, y1 = pcy - 0.5f * ph;
    float x2 = pcx + 0.5f * pw, y2 = pcy + 0.5f * ph;
    x1 = fminf(fmaxf(x1, 0.0f), IMG_SZ);
    y1 = fminf(fmaxf(y1, 0.0f), IMG_SZ);
    x2 = fminf(fmaxf(x2, 0.0f), IMG_SZ);
    y2 = fminf(fmaxf(y2, 0.0f), IMG_SZ);

    const bool valid = ((x2 - x1) >= 1e-3f) && ((y2 - y1) >= 1e-3f);
    const float prob = 1.0f / (1.0f + expf(-logit));

    const int ci = b * M_CAND + ck + slot;
    cand_score[ci] = valid ? prob : -__builtin_inff();
    float4* cb = (float4*)cand_box;
    cb[ci] = make_float4(x1, y1, x2, y2);
}

// ---------------- K3: per-batch sort + greedy NMS, all candidate state in LDS ----------------
__global__ __launch_bounds__(1024)
void rpn_nms(const float* __restrict__ cand_score, const float* __restrict__ cand_box,
             float* __restrict__ out) {
    const int b = blockIdx.x;
    const int tid = threadIdx.x;
    const int nthr = blockDim.x;

    __shared__ float          s_score[SORTN];     // 32 KB
    __shared__ unsigned short s_idx[SORTN];       // 16 KB
    __shared__ float4         s_box[M_CAND];      // ~74.5 KB (level-offset boxes)
    __shared__ unsigned char  s_supp[M_CAND];     // ~4.7 KB
    __shared__ int            s_accept[POST];     // 4 KB
    __shared__ int            s_naccept;

    // ---- async global -> LDS: candidate scores (CDNA5 ASYNCcnt path) ----
    const unsigned long long sb64 = (unsigned long long)(cand_score + (size_t)b * M_CAND);
    for (int p = tid; p < M_CAND; p += nthr) {
        const unsigned lds  = (unsigned)(unsigned long long)(&s_score[p]);  // low 32 bits = LDS offset
        const unsigned goff = (unsigned)(p * 4);
        asm volatile("global_load_async_to_lds_b32 %0, %1, %2"
                     :: "v"(lds), "v"(goff), "s"(sb64) : "memory");
    }
    for (int p = tid; p < SORTN; p += nthr) {
        if (p >= M_CAND) s_score[p] = -__builtin_inff();
        s_idx[p] = (unsigned short)p;
        if (p < M_CAND) s_supp[p] = 0;
    }
    asm volatile("s_wait_asynccnt 0" ::: "memory");
    __syncthreads();

    // ---- bitonic sort: descending score, ascending index on ties ----
    for (int k = 2; k <= SORTN; k <<= 1) {
        for (int j = k >> 1; j > 0; j >>= 1) {
            for (int t = tid; t < SORTN / 2; t += nthr) {
                const int i   = 2 * t - (t & (j - 1));
                const int ixj = i + j;
                const float si = s_score[i], sj = s_score[ixj];
                const int ii = s_idx[i], ij = s_idx[ixj];
                const bool desc = ((i & k) == 0);
                const bool lt = (si < sj) || (si == sj && ii > ij);
                const bool gt = (si > sj) || (si == sj && ii < ij);
                if (desc ? lt : gt) {
                    s_score[i] = sj; s_score[ixj] = si;
                    s_idx[i] = (unsigned short)ij; s_idx[ixj] = (unsigned short)ii;
                }
            }
            __syncthreads();
        }
    }

    // ---- async gather of boxes in sorted order (B128 per candidate) ----
    const unsigned long long bb64 =
        (unsigned long long)cand_box + (unsigned long long)b * M_CAND * 16ull;
    for (int p = tid; p < M_CAND; p += nthr) {
        const unsigned lds  = (unsigned)(unsigned long long)(&s_box[p]);
        const unsigned goff = (unsigned)((int)s_idx[p] * 16);
        asm volatile("global_load_async_to_lds_b128 %0, %1, %2"
                     :: "v"(lds), "v"(goff), "s"(bb64) : "memory");
    }
    asm volatile("s_wait_asynccnt 0" ::: "memory");
    __syncthreads();

    // apply batched_nms per-level offset (level derived from candidate slot)
    for (int p = tid; p < M_CAND; p += nthr) {
        const int orig = s_idx[p];
        const int lvl = (orig < 4000) ? (orig / 1000) : 4;
        const float off = (float)lvl * LVL_OFF;
        float4 bx = s_box[p];
        bx.x += off; bx.y += off; bx.z += off; bx.w += off;
        s_box[p] = bx;
    }
    if (tid == 0) s_naccept = 0;
    __syncthreads();

    // ---- greedy NMS over sorted candidates ----
    for (int p = 0; p < M_CAND; ++p) {
        if (s_naccept >= POST) break;
        const float sp = s_score[p];
        if (!(sp > -__builtin_inff())) break;     // sorted: only -inf remains
        if (s_supp[p]) continue;
        if (tid == 0) { s_accept[s_naccept] = p; s_naccept = s_naccept + 1; }
        const float4 bp = s_box[p];
        const float ap = (bp.z - bp.x) * (bp.w - bp.y);
        for (int j = p + 1 + tid; j < M_CAND; j += nthr) {
            if (s_supp[j]) continue;
            const float4 bj = s_box[j];
            const float ix1 = fmaxf(bp.x, bj.x), iy1 = fmaxf(bp.y, bj.y);
            const float ix2 = fminf(bp.z, bj.z), iy2 = fminf(bp.w, bj.w);
            const float inter = fmaxf(ix2 - ix1, 0.0f) * fmaxf(iy2 - iy1, 0.0f);
            const float aj = (bj.z - bj.x) * (bj.w - bj.y);
            const float iou = inter / (ap + aj - inter + 1e-9f);
            if (iou > NMS_TH_F) s_supp[j] = 1;
        }
        __syncthreads();
    }
    __syncthreads();

    // ---- emit: boxes [8,1000,4] then scores [8,1000] ----
    const int na = s_naccept;
    float* ob = out + (size_t)b * POST * 4;
    float* os = out + (size_t)BATCH * POST * 4 + (size_t)b * POST;
    for (int q = tid; q < POST; q += nthr) {
        if (q < na) {
            const int p = s_accept[q];
            const int orig = s_idx[p];
            const int lvl = (orig < 4000) ? (orig / 1000) : 4;
            const float off = (float)lvl * LVL_OFF;
            const float4 bx = s_box[p];
            ob[q * 4 + 0] = bx.x - off;
            ob[q * 4 + 1] = bx.y - off;
            ob[q * 4 + 2] = bx.z - off;
            ob[q * 4 + 3] = bx.w - off;
            os[q] = s_score[p];
        } else {
            ob[q * 4 + 0] = 0.0f; ob[q * 4 + 1] = 0.0f;
            ob[q * 4 + 2] = 0.0f; ob[q * 4 + 3] = 0.0f;
            os[q] = 0.0f;
        }
    }
}

// ---------------- launch ----------------
extern "C" void kernel_launch(void* const* d_in, const int* in_sizes, int n_in,
                              void* d_out, int out_size, void* d_ws, size_t ws_size,
                              hipStream_t stream) {
    (void)in_sizes; (void)n_in; (void)out_size; (void)ws_size;
    // dict order: scores0, deltas0, scores1, deltas1, ..., scores4, deltas4, anchors
    const float* scores[NLVL];
    const float* deltas[NLVL];
    for (int l = 0; l < NLVL; ++l) {
        scores[l] = (const float*)d_in[2 * l];
        deltas[l] = (const float*)d_in[2 * l + 1];
    }
    const float* anchors = (const float*)d_in[10];

    char* ws = (char*)d_ws;
    float*    cand_score = (float*)(ws + WS_SCORE);
    float*    cand_box   = (float*)(ws + WS_BOX);
    unsigned* thr        = (unsigned*)(ws + WS_THR);
    int*      cntg       = (int*)(ws + WS_CNTG);
    int*      ctrg       = (int*)(ws + WS_CTRG);
    int*      ctre       = (int*)(ws + WS_CTRE);

    rpn_topk_thresh<<<dim3(BATCH * NLVL), 1024, 0, stream>>>(
        scores[0], scores[1], scores[2], scores[3], scores[4], thr, cntg, ctrg, ctre);

    static const int Hs[NLVL]   = {256, 128, 64, 32, 16};
    static const int Ns[NLVL]   = {196608, 49152, 12288, 3072, 768};
    static const int Ks[NLVL]   = {1000, 1000, 1000, 1000, 768};
    static const int CKs[NLVL]  = {0, 1000, 2000, 3000, 4000};
    static const int AOff[NLVL] = {0, 196608, 245760, 258048, 261120};
    for (int l = 0; l < NLVL; ++l) {
        dim3 grid((Ns[l] + 255) / 256, BATCH);
        rpn_select_decode<<<grid, 256, 0, stream>>>(
            scores[l], deltas[l], anchors, Hs[l], Ns[l], Ks[l], l, CKs[l], AOff[l],
            thr, cntg, ctrg, ctre, cand_score, cand_box);
    }

    rpn_nms<<<dim3(BATCH), 1024, 0, stream>>>(cand_score, cand_box, (float*)d_out);
}